// EGAT_26482768347804
// MI455X (gfx1250) — compile-verified
//
#include <hip/hip_runtime.h>
#include <math.h>

#define NHEADS 4

typedef float v2f __attribute__((ext_vector_type(2)));
typedef float v8f __attribute__((ext_vector_type(8)));
typedef unsigned int u32x4 __attribute__((ext_vector_type(4)));
typedef int i32x4 __attribute__((ext_vector_type(4)));
typedef int i32x8 __attribute__((ext_vector_type(8)));

// ---------------------------------------------------------------- utilities
__global__ void fill_f32(float* __restrict__ p, float v, size_t n) {
    size_t i = (size_t)blockIdx.x * blockDim.x + threadIdx.x;
    if (i < n) p[i] = v;
}

__device__ __forceinline__ float sigmoidf_(float x) { return 1.0f / (1.0f + __expf(-x)); }

__device__ __forceinline__ void atomicMaxFloat(float* addr, float val) {
    // IEEE-754 monotonic punning trick: works for mixed-sign values.
    if (val >= 0.0f) atomicMax((int*)addr, __float_as_int(val));
    else             atomicMin((unsigned int*)addr, __float_as_uint(val));
}

// ------------------------------------------------------- TDM 2D tile -> LDS
// Issues one Tensor-Data-Mover DMA of an [nRows x rowLen] f32 tile (row-major,
// row stride = rowLen elements) from global memory into LDS at byte offset
// lds_off, then waits for completion (TENSORcnt==0). D# layout per CDNA5 ISA
// ch.8: group0 = {count=1, lds_addr, global_addr[56:0], type=2},
// group1 = {data_size=4B, tensor_dim0/1, tile_dim0/1, dim0 stride}; groups 2/3
// zero (2D tensor).
__device__ __forceinline__ void tdm_load_tile_f32(unsigned lds_off, const float* gptr,
                                                  int rowLen, int nRows) {
    unsigned long long ga = (unsigned long long)(uintptr_t)gptr;
    u32x4 g0;
    g0.x = 1u;                                                 // count=1 (valid user D#)
    g0.y = lds_off;                                            // lds_addr (bytes)
    g0.z = (unsigned)(ga & 0xFFFFFFFFu);                       // global_addr[31:0]
    g0.w = (unsigned)((ga >> 32) & 0x1FFFFFFu) | (2u << 30);   // global_addr[56:32] | type=2
    i32x8 g1;
    g1[0] = 0x00020000;                                        // data_size = 2 (4 bytes)
    g1[1] = (rowLen & 0xFFFF) << 16;                           // tensor_dim0 (lo16 @ [63:48])
    g1[2] = (nRows & 0xFFFF) << 16;                            // tensor_dim1 (lo16 @ [95:80])
    g1[3] = (rowLen & 0xFFFF) << 16;                           // tile_dim0  @ [127:112]
    g1[4] = (nRows & 0xFFFF);                                  // tile_dim1  @ [143:128]
    g1[5] = rowLen;                                            // tensor_dim0_stride (lo32)
    g1[6] = 0;
    g1[7] = 0;
    i32x4 z4 = {0, 0, 0, 0};
#if defined(__clang_major__) && __clang_major__ >= 23
    i32x8 z8 = {0, 0, 0, 0, 0, 0, 0, 0};
    __builtin_amdgcn_tensor_load_to_lds(g0, g1, z4, z4, z8, 0);
#else
    __builtin_amdgcn_tensor_load_to_lds(g0, g1, z4, z4, 0);
#endif
    __builtin_amdgcn_s_wait_tensorcnt(0);
}

// ---------------------------------------------------------- fp32 WMMA GEMM
// C[M,N] = X[M,K] @ W[K,N] (+ bias).  M%16==0, N%16==0, K%4==0.
// One wave computes one 16x16 tile of C via V_WMMA_F32_16X16X4_F32.
// A fragment (16x4 f32): lane = M + 16*(K/2), reg = K&1.
// B fragment (4x16 f32): K = 2*(lane>=16)+reg, N = lane&15.
// C/D (16x16 f32): reg v, lane l -> M = v + 8*(l>=16), N = l&15.
__global__ void gemm_wmma_f32(const float* __restrict__ X, const float* __restrict__ W,
                              const float* __restrict__ bias, float* __restrict__ C,
                              int M, int N, int K) {
    const int lane = threadIdx.x & 31;
    const int wave = threadIdx.x >> 5;
    const int tilesN = N >> 4;
    const int totalTiles = (M >> 4) * tilesN;
    const int tile = blockIdx.x * 8 + wave;          // 8 waves / 256-thread block
    if (tile >= totalTiles) return;                  // wave-uniform: EXEC stays all-1 for WMMA
    const int tm = (tile / tilesN) << 4;
    const int tn = (tile % tilesN) << 4;

    const int half = lane >> 4;    // 0 | 1
    const int l16  = lane & 15;

    v8f acc = {};
    const float* xrow = X + (size_t)(tm + l16) * K + 2 * half;   // element k = k0 + 2*half (+1)
    const float* wcol = W + (size_t)(2 * half) * N + tn + l16;
    for (int k0 = 0; k0 < K; k0 += 4) {
        v2f a, b;
        a.x = xrow[k0];
        a.y = xrow[k0 + 1];
        b.x = wcol[(size_t)k0 * N];
        b.y = wcol[(size_t)(k0 + 1) * N];
        acc = __builtin_amdgcn_wmma_f32_16x16x4_f32(false, a, false, b,
                                                    (short)0, acc, false, false);
    }
    const float bv = bias ? bias[tn + l16] : 0.0f;
    #pragma unroll
    for (int v = 0; v < 8; ++v) {
        const int m = v + 8 * half;
        C[(size_t)(tm + m) * N + tn + l16] = acc[v] + bv;
    }
}

// -------------------------------- fused edge GEMM (TDM-staged A) + combine
// f[e,:] = leaky_relu( (ef @ Wfij)[e,:] + Pi[src[e],:] + Pj[dst[e],:] + eb, 0.01 )
// One block owns one 16-edge row of output tiles (tilesN = HD/16 waves). The
// streaming A tile (16 x K of ef, <= 4KB) is DMA'd once per block into LDS by
// the Tensor Data Mover (wave 0 issues + s_wait_tensorcnt, barrier releases the
// block), so ef is read from HBM exactly once and A fragments come from LDS.
// W stays in L1 (tiny). The 16x16 Pf tile never touches memory: gathers + bias
// + activation are applied in the WMMA epilogue, saving a full write+read of
// the [E, H*doe] tensor (the largest stream in the pipeline).
__global__ void edge_gemm_fused(const float* __restrict__ X,     // ef_cur [E,K]
                                const float* __restrict__ W,     // Wfij [K,HD]
                                const float* __restrict__ Pi,    // [Nnodes,HD]
                                const float* __restrict__ Pj,    // [Nnodes,HD]
                                const float* __restrict__ eb,    // [HD]
                                const int* __restrict__ src, const int* __restrict__ dst,
                                float* __restrict__ f,           // [E,HD]
                                int E, int HD, int K) {
    __shared__ float Ash[16 * 64];                   // 16 rows x K (K <= 64)

    const int lane = threadIdx.x & 31;
    const int wave = threadIdx.x >> 5;
    const int tm = blockIdx.x << 4;                  // first edge of this block
    const int tn = wave << 4;                        // this wave's 16 output columns

    if (wave == 0) {
        tdm_load_tile_f32((unsigned)(uintptr_t)(void*)Ash,
                          X + (size_t)tm * K, K, 16);
    }
    __syncthreads();

    const int half = lane >> 4;
    const int l16  = lane & 15;

    // Row indices for the 8 output rows this lane contributes to (uniform per half).
    int se[8], de[8];
    #pragma unroll
    for (int v = 0; v < 8; ++v) {
        const int e = tm + v + 8 * half;
        se[v] = src[e];
        de[v] = dst[e];
        __builtin_prefetch(&Pi[(size_t)se[v] * HD + tn + l16], 0, 0);
        __builtin_prefetch(&Pj[(size_t)de[v] * HD + tn + l16], 0, 0);
    }

    v8f acc = {};
    const float* arow = Ash + l16 * K + 2 * half;    // LDS-resident A fragments
    const float* wcol = W + (size_t)(2 * half) * HD + tn + l16;
    for (int k0 = 0; k0 < K; k0 += 4) {
        v2f a, b;
        a.x = arow[k0];
        a.y = arow[k0 + 1];
        b.x = wcol[(size_t)k0 * HD];
        b.y = wcol[(size_t)(k0 + 1) * HD];
        acc = __builtin_amdgcn_wmma_f32_16x16x4_f32(false, a, false, b,
                                                    (short)0, acc, false, false);
    }

    const float bb = eb[tn + l16];
    #pragma unroll
    for (int v = 0; v < 8; ++v) {
        const int e = tm + v + 8 * half;
        float val = acc[v] + Pi[(size_t)se[v] * HD + tn + l16]
                           + Pj[(size_t)de[v] * HD + tn + l16] + bb;
        f[(size_t)e * HD + tn + l16] = val > 0.0f ? val : 0.01f * val;
    }
}

// -------------------------------------------------------------- edge kernels
// s[e,h] = sum_d f[e,h,d]*attn[h,d]; atomic segment-max into node_max[dst,h].
__global__ void edge_attn(const float* __restrict__ f, const float* __restrict__ attn,
                          const int* __restrict__ dst, float* __restrict__ eA,
                          float* __restrict__ node_max, int E, int doe) {
    int idx = blockIdx.x * blockDim.x + threadIdx.x;   // e*NHEADS + h
    if (idx >= E * NHEADS) return;
    int e = idx / NHEADS, h = idx % NHEADS;
    const float* fp = f + (size_t)e * NHEADS * doe + (size_t)h * doe;
    const float* ap = attn + h * doe;
    float s = 0.0f;
    for (int d = 0; d < doe; ++d) s += fp[d] * ap[d];
    eA[idx] = s;
    atomicMaxFloat(&node_max[dst[e] * NHEADS + h], s);
}

// ee = exp(s - max[dst]); atomic segment-sum into den[dst,h]. eA updated in place.
__global__ void edge_exp(float* __restrict__ eA, const float* __restrict__ node_max,
                         float* __restrict__ den, const int* __restrict__ dst, int E) {
    int idx = blockIdx.x * blockDim.x + threadIdx.x;
    if (idx >= E * NHEADS) return;
    int e = idx / NHEADS, h = idx % NHEADS;
    float ee = __expf(eA[idx] - node_max[dst[e] * NHEADS + h]);
    eA[idx] = ee;
    atomicAdd(&den[dst[e] * NHEADS + h], ee);
}

// hout[dst,h,:] += Ft[src,h,:] * (ee/den[dst,h])
__global__ void edge_aggregate(const float* __restrict__ eA, const float* __restrict__ den,
                               const float* __restrict__ Ft, const int* __restrict__ src,
                               const int* __restrict__ dst, float* __restrict__ hout,
                               int E, int don) {
    int idx = blockIdx.x * blockDim.x + threadIdx.x;
    if (idx >= E * NHEADS) return;
    int e = idx / NHEADS, h = idx % NHEADS;
    int s = src[e], d = dst[e];
    float a = eA[idx] / den[d * NHEADS + h];
    const float* ftp = Ft + (size_t)s * NHEADS * don + (size_t)h * don;
    float* ho = hout + (size_t)d * NHEADS * don + (size_t)h * don;
    for (int k = 0; k < don; ++k) atomicAdd(&ho[k], ftp[k] * a);
}

// nf[n,d] = mean_h hout[n,h,d]; optional eval-BN; optional residual add.
__global__ void node_finalize(const float* __restrict__ hout, float* __restrict__ nf,
                              const float* __restrict__ g, const float* __restrict__ b,
                              const float* __restrict__ rm, const float* __restrict__ rv,
                              const float* __restrict__ resid, int N, int don) {
    int idx = blockIdx.x * blockDim.x + threadIdx.x;   // n*don + d
    if (idx >= N * don) return;
    int n = idx / don, d = idx % don;
    float s = 0.0f;
    for (int h = 0; h < NHEADS; ++h) s += hout[(size_t)n * NHEADS * don + (size_t)h * don + d];
    s *= (1.0f / NHEADS);
    if (g)     s = (s - rm[d]) * rsqrtf(rv[d] + 1e-5f) * g[d] + b[d];
    if (resid) s += resid[idx];
    nf[idx] = s;
}

// ef[e,d] = mean_h f[e,h,d]
__global__ void edge_mean(const float* __restrict__ f, float* __restrict__ ef, int E, int doe) {
    size_t idx = (size_t)blockIdx.x * blockDim.x + threadIdx.x;   // e*doe + d
    if (idx >= (size_t)E * doe) return;
    size_t e = idx / doe;
    int d = (int)(idx % doe);
    float s = 0.0f;
    for (int h = 0; h < NHEADS; ++h) s += f[e * NHEADS * doe + (size_t)h * doe + d];
    ef[idx] = s * (1.0f / NHEADS);
}

// ---------------------------------------------------------------- Set2Set
// Single 1024-thread workgroup; LSTM state in LDS; 6 sequential iterations.
__global__ void set2set_kernel(const float* __restrict__ feat,
                               const float* __restrict__ Wih0, const float* __restrict__ Whh0,
                               const float* __restrict__ bih0, const float* __restrict__ bhh0,
                               const float* __restrict__ Wih1, const float* __restrict__ Whh1,
                               const float* __restrict__ bih1, const float* __restrict__ bhh1,
                               float* __restrict__ scores, float* __restrict__ out, int N) {
    __shared__ float qs[128];
    __shared__ float h0[64], c0[64], h1[64], c1[64];
    __shared__ float gates[256];
    __shared__ float red[1024];
    __shared__ float r_sh[64];
    __shared__ float den_sh;

    const int tid = threadIdx.x;
    if (tid < 128) qs[tid] = 0.0f;
    if (tid < 64) { h0[tid] = 0.0f; c0[tid] = 0.0f; h1[tid] = 0.0f; c1[tid] = 0.0f; }
    __syncthreads();

    for (int it = 0; it < 6; ++it) {
        // LSTM layer 0: gates = Wih0 @ q_star + Whh0 @ h0 + biases  (gate order i,f,g,o)
        if (tid < 256) {
            float gv = bih0[tid] + bhh0[tid];
            const float* wi = Wih0 + tid * 128;
            for (int t = 0; t < 128; ++t) gv += wi[t] * qs[t];
            const float* wh = Whh0 + tid * 64;
            for (int t = 0; t < 64; ++t) gv += wh[t] * h0[t];
            gates[tid] = gv;
        }
        __syncthreads();
        if (tid < 64) {
            float ig = sigmoidf_(gates[tid]);
            float fg = sigmoidf_(gates[64 + tid]);
            float gg = tanhf(gates[128 + tid]);
            float og = sigmoidf_(gates[192 + tid]);
            float c = fg * c0[tid] + ig * gg;
            c0[tid] = c;
            h0[tid] = og * tanhf(c);
        }
        __syncthreads();
        // LSTM layer 1
        if (tid < 256) {
            float gv = bih1[tid] + bhh1[tid];
            const float* wi = Wih1 + tid * 64;
            for (int t = 0; t < 64; ++t) gv += wi[t] * h0[t];
            const float* wh = Whh1 + tid * 64;
            for (int t = 0; t < 64; ++t) gv += wh[t] * h1[t];
            gates[tid] = gv;
        }
        __syncthreads();
        if (tid < 64) {
            float ig = sigmoidf_(gates[tid]);
            float fg = sigmoidf_(gates[64 + tid]);
            float gg = tanhf(gates[128 + tid]);
            float og = sigmoidf_(gates[192 + tid]);
            float c = fg * c1[tid] + ig * gg;
            c1[tid] = c;
            float h = og * tanhf(c);
            h1[tid] = h;
            qs[tid] = h;                 // q part of q_star
        }
        __syncthreads();

        // scores[n] = feat[n] . q ; block max
        float lmax = -INFINITY;
        for (int n = tid; n < N; n += 1024) {
            const float* fp = feat + (size_t)n * 64;
            float s = 0.0f;
            for (int d = 0; d < 64; ++d) s += fp[d] * h1[d];
            scores[n] = s;
            lmax = fmaxf(lmax, s);
        }
        red[tid] = lmax;
        __syncthreads();
        for (int off = 512; off > 0; off >>= 1) {
            if (tid < off) red[tid] = fmaxf(red[tid], red[tid + off]);
            __syncthreads();
        }
        float smax = red[0];
        __syncthreads();

        // den = sum exp; r = sum exp*feat
        float rl[64];
        #pragma unroll
        for (int d = 0; d < 64; ++d) rl[d] = 0.0f;
        float dl = 0.0f;
        for (int n = tid; n < N; n += 1024) {
            float e = __expf(scores[n] - smax);
            dl += e;
            const float* fp = feat + (size_t)n * 64;
            for (int d = 0; d < 64; ++d) rl[d] += e * fp[d];
        }
        red[tid] = dl;
        if (tid < 64) r_sh[tid] = 0.0f;
        __syncthreads();
        for (int off = 512; off > 0; off >>= 1) {
            if (tid < off) red[tid] += red[tid + off];
            __syncthreads();
        }
        if (tid == 0) den_sh = red[0];
        for (int d = 0; d < 64; ++d) atomicAdd(&r_sh[d], rl[d]);
        __syncthreads();
        if (tid < 64) qs[64 + tid] = r_sh[tid] / den_sh;   // r part of q_star
        __syncthreads();
    }
    if (tid < 128) out[tid] = qs[tid];
}

// ---------------------------------------------------------------- launcher
extern "C" void kernel_launch(void* const* d_in, const int* in_sizes, int n_in,
                              void* d_out, int out_size, void* d_ws, size_t ws_size,
                              hipStream_t stream) {
    (void)n_in; (void)out_size; (void)ws_size;
    const float* node_feat = (const float*)d_in[0];
    const float* edge_feat = (const float*)d_in[1];
    const int*   src       = (const int*)d_in[2];
    const int*   dst       = (const int*)d_in[3];
    const int N = in_sizes[0] / 64;
    const int E = in_sizes[1] / 64;

    const int din_n[4] = {64, 64, 32, 16};
    const int din_e[4] = {64, 64, 32, 16};
    const int don[4]   = {64, 32, 16, 64};
    const int doe[4]   = {64, 32, 16, 64};

    const float *Wn[4], *bn[4], *Wni[4], *Wnj[4], *Wfij[4], *attn[4], *eb[4];
    for (int l = 0; l < 4; ++l) {
        int b = 4 + 7 * l;
        Wn[l]   = (const float*)d_in[b + 0];
        bn[l]   = (const float*)d_in[b + 1];
        Wni[l]  = (const float*)d_in[b + 2];
        Wnj[l]  = (const float*)d_in[b + 3];
        Wfij[l] = (const float*)d_in[b + 4];
        attn[l] = (const float*)d_in[b + 5];
        eb[l]   = (const float*)d_in[b + 6];
    }
    const float *bng[3], *bnb[3], *bnrm[3], *bnrv[3];
    for (int i = 0; i < 3; ++i) {
        int b = 32 + 4 * i;
        bng[i]  = (const float*)d_in[b + 0];
        bnb[i]  = (const float*)d_in[b + 1];
        bnrm[i] = (const float*)d_in[b + 2];
        bnrv[i] = (const float*)d_in[b + 3];
    }
    const float *Wih[2], *Whh[2], *bih[2], *bhh[2];
    for (int k = 0; k < 2; ++k) {
        int b = 44 + 4 * k;
        Wih[k] = (const float*)d_in[b + 0];
        Whh[k] = (const float*)d_in[b + 1];
        bih[k] = (const float*)d_in[b + 2];
        bhh[k] = (const float*)d_in[b + 3];
    }

    // ---- workspace layout (floats) ----
    float* ws = (float*)d_ws;
    size_t o = 0;
    float* Pi  = ws + o; o += (size_t)N * 256;
    float* Pj  = ws + o; o += (size_t)N * 256;
    float* Ft  = ws + o; o += (size_t)N * 256;
    float* Ho  = ws + o; o += (size_t)N * 256;
    float* Fe  = ws + o; o += (size_t)E * 256;     // fused edge-GEMM output f_out
    float* EA  = ws + o; o += (size_t)E * NHEADS;  // scores -> ee in place
    float* NM  = ws + o; o += (size_t)N * NHEADS;
    float* DEN = ws + o; o += (size_t)N * NHEADS;
    float* nfA = ws + o; o += (size_t)N * 64;
    float* nfB = ws + o; o += (size_t)N * 64;
    float* efA = ws + o; o += (size_t)E * 64;
    float* efB = ws + o; o += (size_t)E * 64;
    float* SC  = ws + o; o += (size_t)N;

    const float* nf_cur = node_feat;
    const float* ef_cur = edge_feat;
    float* nf_next[4] = {nfA, nfB, nfA, nfB};
    float* ef_next[4] = {efA, efB, efA, nullptr};

    auto gblk = [](size_t n) { return (unsigned)((n + 255) / 256); };

    for (int l = 0; l < 4; ++l) {
        const int HD  = NHEADS * doe[l];
        const int HDn = NHEADS * don[l];

        // init segmented-reduction buffers (workspace is poisoned, so every call)
        fill_f32<<<gblk((size_t)N * NHEADS), 256, 0, stream>>>(NM, -INFINITY, (size_t)N * NHEADS);
        fill_f32<<<gblk((size_t)N * NHEADS), 256, 0, stream>>>(DEN, 0.0f, (size_t)N * NHEADS);
        fill_f32<<<gblk((size_t)N * HDn), 256, 0, stream>>>(Ho, 0.0f, (size_t)N * HDn);

        // node projections (fp32 WMMA)
        {
            unsigned t;
            t = (unsigned)(((size_t)(N / 16) * (HD / 16) + 7) / 8);
            gemm_wmma_f32<<<t, 256, 0, stream>>>(nf_cur, Wni[l], nullptr, Pi, N, HD, din_n[l]);
            gemm_wmma_f32<<<t, 256, 0, stream>>>(nf_cur, Wnj[l], nullptr, Pj, N, HD, din_n[l]);
            t = (unsigned)(((size_t)(N / 16) * (HDn / 16) + 7) / 8);
            gemm_wmma_f32<<<t, 256, 0, stream>>>(nf_cur, Wn[l], bn[l], Ft, N, HDn, din_n[l]);
        }

        // fused edge projection (TDM-staged A) + gathers + bias + leaky-relu
        {
            unsigned blocks = (unsigned)(E / 16);          // one block per 16-edge tile row
            unsigned threads = (unsigned)(32 * (HD / 16)); // one wave per 16-col tile
            edge_gemm_fused<<<blocks, threads, 0, stream>>>(ef_cur, Wfij[l], Pi, Pj, eb[l],
                                                            src, dst, Fe, E, HD, din_e[l]);
        }

        edge_attn<<<gblk((size_t)E * NHEADS), 256, 0, stream>>>(Fe, attn[l], dst, EA, NM, E, doe[l]);
        edge_exp<<<gblk((size_t)E * NHEADS), 256, 0, stream>>>(EA, NM, DEN, dst, E);
        edge_aggregate<<<gblk((size_t)E * NHEADS), 256, 0, stream>>>(EA, DEN, Ft, src, dst, Ho, E, don[l]);

        const float* g  = (l < 3) ? bng[l]  : nullptr;
        const float* bb = (l < 3) ? bnb[l]  : nullptr;
        const float* rm = (l < 3) ? bnrm[l] : nullptr;
        const float* rv = (l < 3) ? bnrv[l] : nullptr;
        const float* resid = (l == 3) ? node_feat : nullptr;
        node_finalize<<<gblk((size_t)N * don[l]), 256, 0, stream>>>(Ho, nf_next[l], g, bb, rm, rv,
                                                                    resid, N, don[l]);
        if (l < 3) {
            edge_mean<<<gblk((size_t)E * doe[l]), 256, 0, stream>>>(Fe, ef_next[l], E, doe[l]);
            ef_cur = ef_next[l];
        }
        nf_cur = nf_next[l];
    }

    set2set_kernel<<<1, 1024, 0, stream>>>(nf_cur, Wih[0], Whh[0], bih[0], bhh[0],
                                           Wih[1], Whh[1], bih[1], bhh[1],
                                           SC, (float*)d_out, N);
}